// LightSAGEConv_42511586296448
// MI455X (gfx1250) — compile-verified
//
#include <hip/hip_runtime.h>
#include <hip/hip_bf16.h>
#include <stdint.h>

typedef float v2f __attribute__((ext_vector_type(2)));
typedef float v8f __attribute__((ext_vector_type(8)));

#define N_NODES 100000
#define CFEAT   64

// Guaranteed hardware f32 atomic add (no CAS loop), device scope so the
// following kernel sees the reductions.
__device__ __forceinline__ void gatomic_add_f32(float* p, float v) {
    asm volatile("global_atomic_add_f32 %0, %1, off scope:SCOPE_DEV"
                 :: "v"(p), "v"(v) : "memory");
}

// ---------------- workspace zeroing (agg + cnt) ----------------
__global__ void zero_ws_kernel(float* __restrict__ p, int n) {
    int i = blockIdx.x * blockDim.x + threadIdx.x;
    if (i < n) p[i] = 0.0f;
}

// ---------------- fold weights: Wc = linr_w @ lin_w, b2 = linr_w @ lin_b ----
__global__ void fold_weights_kernel(const float* __restrict__ lin_w,
                                    const float* __restrict__ lin_b,
                                    const float* __restrict__ linr_w,
                                    float* __restrict__ Wc,
                                    float* __restrict__ b2) {
    int t = blockIdx.x * blockDim.x + threadIdx.x;
    if (t < CFEAT * CFEAT) {
        int i = t >> 6, k = t & 63;
        float s = 0.f;
        #pragma unroll 8
        for (int j = 0; j < CFEAT; ++j) s += linr_w[i * CFEAT + j] * lin_w[j * CFEAT + k];
        Wc[t] = s;
    } else if (t < CFEAT * CFEAT + CFEAT) {
        int i = t - CFEAT * CFEAT;
        float s = 0.f;
        #pragma unroll 8
        for (int j = 0; j < CFEAT; ++j) s += linr_w[i * CFEAT + j] * lin_b[j];
        b2[i] = s;
    }
}

// ---------------- edge scatter: agg[dst] += x[src], cnt[dst] += 1 ----------
// One wave per edge; each lane handles 2 contiguous channels (64 ch / 32 lanes).
__global__ void edge_scatter_kernel(const float* __restrict__ x,
                                    const long long* __restrict__ ei,  // [2,E]
                                    float* __restrict__ agg,
                                    float* __restrict__ cnt,
                                    int E) {
    int gid  = blockIdx.x * blockDim.x + threadIdx.x;
    int e    = gid >> 5;
    int lane = gid & 31;
    if (e >= E) return;
    long long src = ei[e];
    long long dst = ei[(long long)E + e];
    const float2 xv = *(const float2*)(x + (size_t)src * CFEAT + lane * 2);
    float* dp = agg + (size_t)dst * CFEAT + lane * 2;
    gatomic_add_f32(dp,     xv.x);
    gatomic_add_f32(dp + 1, xv.y);
    if (lane == 0) gatomic_add_f32(cnt + dst, 1.0f);
}

// ---------------- fused epilogue via V_WMMA_F32_16X16X4_F32 ----------------
// out = x @ linl_w^T + linl_b + (agg/max(cnt,1)) @ Wc^T + min(cnt,1)*b2
// Weights staged once per block into LDS; B-fragments served by ds_load_b64.
// One wave per 16-node tile; 4 f32 accumulators (16x16) cover all 64 columns.
__global__ __launch_bounds__(256)
void sage_out_kernel(const float* __restrict__ x,
                     const float* __restrict__ linl_w,
                     const float* __restrict__ linl_b,
                     const float* __restrict__ agg,
                     const float* __restrict__ cnt,
                     const float* __restrict__ Wc,
                     const float* __restrict__ b2,
                     float* __restrict__ out) {
    __shared__ float s_lw[CFEAT * CFEAT];   // 16 KB
    __shared__ float s_wc[CFEAT * CFEAT];   // 16 KB
    __shared__ float s_lb[CFEAT];
    __shared__ float s_b2[CFEAT];

    const int tid  = threadIdx.x;
    const int lane = tid & 31;
    const int wave = tid >> 5;
    const int tile = blockIdx.x * 8 + wave;

    // ---- stage weights to LDS (all 256 threads, 4x float4 per matrix) ----
    #pragma unroll
    for (int i = tid; i < (CFEAT * CFEAT) / 4; i += 256) {
        ((float4*)s_lw)[i] = ((const float4*)linl_w)[i];
        ((float4*)s_wc)[i] = ((const float4*)Wc)[i];
    }
    if (tid < CFEAT)                 s_lb[tid]          = linl_b[tid];
    else if (tid < 2 * CFEAT)        s_b2[tid - CFEAT]  = b2[tid - CFEAT];
    __syncthreads();

    if (tile < N_NODES / 16) {              // wave-uniform: EXEC all-1s inside
        const int node0 = tile * 16;
        const int l16   = lane & 15;
        const int hi    = lane >> 4;

        // A-fragment row (lanes 0-15 and 16-31 both map M=0..15, K split by hi)
        const int   rowA = node0 + l16;
        const float invA = __builtin_amdgcn_rcpf(fmaxf(cnt[rowA], 1.0f));

        // D-fragment rows: VGPR r holds row node0 + r + 8*hi
        float c1m[8];
        #pragma unroll
        for (int r = 0; r < 8; ++r)
            c1m[r] = fminf(cnt[node0 + r + 8 * hi], 1.0f);   // min(cnt,1), cnt>=0

        v8f acc[4];
        #pragma unroll
        for (int nt = 0; nt < 4; ++nt) {
            const int n = nt * 16 + l16;    // column owned by this lane
            const float bl = s_lb[n];
            const float br = s_b2[n];
            #pragma unroll
            for (int r = 0; r < 8; ++r) acc[nt][r] = bl + c1m[r] * br;
        }

        const float* xrow = x   + (size_t)rowA * CFEAT;
        const float* arow = agg + (size_t)rowA * CFEAT;

        #pragma unroll
        for (int kb = 0; kb < CFEAT; kb += 4) {
            const int kk = kb + 2 * hi;     // this lane's K slice
            v2f aX = { xrow[kk], xrow[kk + 1] };
            v2f aM = { arow[kk] * invA, arow[kk + 1] * invA };
            #pragma unroll
            for (int nt = 0; nt < 4; ++nt) {
                const int n = nt * 16 + l16;
                v2f bL = *(const v2f*)(&s_lw[n * CFEAT + kk]);  // ds_load_b64
                v2f bC = *(const v2f*)(&s_wc[n * CFEAT + kk]);  // ds_load_b64
                acc[nt] = __builtin_amdgcn_wmma_f32_16x16x4_f32(
                    false, aX, false, bL, (short)0, acc[nt], false, false);
                acc[nt] = __builtin_amdgcn_wmma_f32_16x16x4_f32(
                    false, aM, false, bC, (short)0, acc[nt], false, false);
            }
        }

        #pragma unroll
        for (int nt = 0; nt < 4; ++nt) {
            const int n = nt * 16 + l16;
            #pragma unroll
            for (int r = 0; r < 8; ++r)
                out[(size_t)(node0 + r + 8 * hi) * CFEAT + n] = acc[nt][r];
        }
    }
}

extern "C" void kernel_launch(void* const* d_in, const int* in_sizes, int n_in,
                              void* d_out, int out_size, void* d_ws, size_t ws_size,
                              hipStream_t stream) {
    const float*     x      = (const float*)d_in[0];
    const long long* ei     = (const long long*)d_in[1];   // int64 per reference
    const float*     lin_w  = (const float*)d_in[2];
    const float*     lin_b  = (const float*)d_in[3];
    const float*     linl_w = (const float*)d_in[4];
    const float*     linl_b = (const float*)d_in[5];
    const float*     linr_w = (const float*)d_in[6];
    float*           out    = (float*)d_out;

    // workspace layout: agg[N*64] | cnt[N] | Wc[64*64] | b2[64]
    float* agg = (float*)d_ws;
    float* cnt = agg + (size_t)N_NODES * CFEAT;
    float* Wc  = cnt + N_NODES;
    float* b2  = Wc + CFEAT * CFEAT;

    const int E = in_sizes[1] / 2;

    const int nz = N_NODES * CFEAT + N_NODES;
    zero_ws_kernel<<<(nz + 255) / 256, 256, 0, stream>>>(agg, nz);

    fold_weights_kernel<<<(CFEAT * CFEAT + CFEAT + 255) / 256, 256, 0, stream>>>(
        lin_w, lin_b, linr_w, Wc, b2);

    const long long sthreads = (long long)E * 32;
    edge_scatter_kernel<<<(int)((sthreads + 255) / 256), 256, 0, stream>>>(
        x, ei, agg, cnt, E);

    const int ntiles = N_NODES / 16;
    sage_out_kernel<<<(ntiles + 7) / 8, 256, 0, stream>>>(
        x, linl_w, linl_b, agg, cnt, Wc, b2, out);
}